// SyntaxGCN_31868657336593
// MI455X (gfx1250) — compile-verified
//
#include <hip/hip_runtime.h>

typedef float v2f __attribute__((ext_vector_type(2)));
typedef float v8f __attribute__((ext_vector_type(8)));

#define NF 32  // hidden feature width

// ---------------- zero accumulators ----------------
__global__ void zero_kernel(float* __restrict__ p, int n) {
    int i = blockIdx.x * blockDim.x + threadIdx.x;
    int stride = gridDim.x * blockDim.x;
    for (; i < n; i += stride) p[i] = 0.0f;
}

// ---------------- degree over dst ----------------
__global__ void degree_kernel(const int* __restrict__ dst, float* __restrict__ deg, int n_edges) {
    int e = blockIdx.x * blockDim.x + threadIdx.x;
    if (e < n_edges) atomicAdd(&deg[dst[e]], 1.0f);
}

__global__ void rsqrt_kernel(float* __restrict__ dinv, int n_nodes) {
    int i = blockIdx.x * blockDim.x + threadIdx.x;
    if (i < n_nodes) dinv[i] = rsqrtf(dinv[i] + 1.0f);
}

// ---------------- xw1 = x @ W1  (K=3, scalar FMAs) ----------------
__global__ void xw1_kernel(const float* __restrict__ x, const float* __restrict__ W1,
                           float* __restrict__ xw, int n_nodes) {
    int tid = blockIdx.x * blockDim.x + threadIdx.x;
    int i = tid >> 2;
    if (i >= n_nodes) return;
    int c = (tid & 3) * 8;
    float x0 = x[i * 3 + 0], x1 = x[i * 3 + 1], x2 = x[i * 3 + 2];
#pragma unroll
    for (int j = 0; j < 8; ++j) {
        int f = c + j;
        xw[(size_t)i * NF + f] = fmaf(x0, W1[f], fmaf(x1, W1[32 + f], x2 * W1[64 + f]));
    }
}

// ---------------- edge scatter: agg[dst] += xw[src]*norm ----------------
__global__ void scatter_kernel(const int* __restrict__ src, const int* __restrict__ dst,
                               const float* __restrict__ dinv, const float* __restrict__ xw,
                               float* __restrict__ agg, int n_edges) {
    int tid = blockIdx.x * blockDim.x + threadIdx.x;
    int e = tid >> 3;
    if (e >= n_edges) return;
    int c = (tid & 7) * 4;
    int s = src[e], d = dst[e];
    float nrm = dinv[s] * dinv[d];
    float4 v = *(const float4*)(xw + (size_t)s * NF + c);
    float* o = agg + (size_t)d * NF + c;
    atomicAdd(o + 0, v.x * nrm);
    atomicAdd(o + 1, v.y * nrm);
    atomicAdd(o + 2, v.z * nrm);
    atomicAdd(o + 3, v.w * nrm);
}

// ---------------- finalize layer: h = relu(agg + xw*dinv^2 + b); agg := 0 ----------------
__global__ void finalize1_kernel(float* __restrict__ agg, float* __restrict__ xw,
                                 const float* __restrict__ dinv, const float* __restrict__ b,
                                 int n_nodes) {
    int tid = blockIdx.x * blockDim.x + threadIdx.x;
    int i = tid >> 3;
    if (i >= n_nodes) return;
    int c = (tid & 7) * 4;
    float s = dinv[i]; s *= s;
    size_t idx = (size_t)i * NF + c;
    float4 a = *(const float4*)(agg + idx);
    float4 w = *(const float4*)(xw + idx);
    float4 r;
    r.x = fmaxf(fmaf(w.x, s, a.x) + b[c + 0], 0.0f);
    r.y = fmaxf(fmaf(w.y, s, a.y) + b[c + 1], 0.0f);
    r.z = fmaxf(fmaf(w.z, s, a.z) + b[c + 2], 0.0f);
    r.w = fmaxf(fmaf(w.w, s, a.w) + b[c + 3], 0.0f);
    *(float4*)(xw + idx) = r;                       // h overwrites xw (elementwise-safe)
    *(float4*)(agg + idx) = make_float4(0.f, 0.f, 0.f, 0.f);  // re-zero for layer 2
}

// ---------------- xw2 = h1 @ W2 via V_WMMA_F32_16X16X4_F32 ----------------
// One wave computes a 16x32 output stripe (two 16x16 C tiles), K=32 in 8 steps of 4.
// A layout (16x4 f32): lanes 0-15 hold M=0..15 K={k,k+1}; lanes 16-31 K={k+2,k+3}.
// B layout (4x16):     N striped across lanes; K rows split across vgpr/lane-halves.
__global__ void __launch_bounds__(256) gemm_wmma_kernel(const float* __restrict__ H,
                                                        const float* __restrict__ W,
                                                        float* __restrict__ O, int n_nodes) {
    __shared__ float Ws[NF * NF];
    {   // 256 threads x float4 = 1024 floats: stage W2 once per block
        int t = threadIdx.x;
        ((float4*)Ws)[t] = ((const float4*)W)[t];
    }
    __syncthreads();

    int wave = (blockIdx.x * blockDim.x + threadIdx.x) >> 5;
    int lane = threadIdx.x & 31;
    int m0 = wave * 16;
    if (m0 < n_nodes) {  // wave-uniform predicate: EXEC all-ones inside
        int nidx  = lane & 15;
        int khalf = (lane >> 4) << 1;  // 0 or 2
        v8f c0 = {};
        v8f c1 = {};
        const float* hrow = H + (size_t)(m0 + nidx) * NF;
#pragma unroll
        for (int k = 0; k < 8; ++k) {
            int kk = k * 4 + khalf;
            v2f a;  a.x  = hrow[kk];                 a.y  = hrow[kk + 1];
            v2f b0; b0.x = Ws[kk * NF + nidx];       b0.y = Ws[(kk + 1) * NF + nidx];
            v2f b1; b1.x = Ws[kk * NF + 16 + nidx];  b1.y = Ws[(kk + 1) * NF + 16 + nidx];
            c0 = __builtin_amdgcn_wmma_f32_16x16x4_f32(false, a, false, b0, (short)0, c0, false, false);
            c1 = __builtin_amdgcn_wmma_f32_16x16x4_f32(false, a, false, b1, (short)0, c1, false, false);
        }
        // D layout: vgpr r -> row m0+r (lanes 0-15) / m0+8+r (lanes 16-31), col = nidx
#pragma unroll
        for (int r = 0; r < 8; ++r) {
            int m = m0 + r + ((lane >> 4) << 3);
            O[(size_t)m * NF + nidx]      = c0[r];
            O[(size_t)m * NF + 16 + nidx] = c1[r];
        }
    }
}

// ---------------- finalize layer 2 fused with global mean pool ----------------
__global__ void finalize2_pool_kernel(const float* __restrict__ agg, const float* __restrict__ xw,
                                      const float* __restrict__ dinv, const float* __restrict__ b,
                                      const int* __restrict__ batch,
                                      float* __restrict__ psum, float* __restrict__ pcnt,
                                      int n_nodes) {
    int tid = blockIdx.x * blockDim.x + threadIdx.x;
    int i = tid >> 3;
    if (i >= n_nodes) return;
    int c = (tid & 7) * 4;
    float s = dinv[i]; s *= s;
    size_t idx = (size_t)i * NF + c;
    float4 a = *(const float4*)(agg + idx);
    float4 w = *(const float4*)(xw + idx);
    float4 h;
    h.x = fmaxf(fmaf(w.x, s, a.x) + b[c + 0], 0.0f);
    h.y = fmaxf(fmaf(w.y, s, a.y) + b[c + 1], 0.0f);
    h.z = fmaxf(fmaf(w.z, s, a.z) + b[c + 2], 0.0f);
    h.w = fmaxf(fmaf(w.w, s, a.w) + b[c + 3], 0.0f);
    int g = batch[i];
    float* o = psum + (size_t)g * NF + c;
    atomicAdd(o + 0, h.x);
    atomicAdd(o + 1, h.y);
    atomicAdd(o + 2, h.z);
    atomicAdd(o + 3, h.w);
    if (c == 0) atomicAdd(&pcnt[g], 1.0f);
}

// ---------------- head: out = (psum/cnt) @ Wl + bl ----------------
__global__ void head_kernel(const float* __restrict__ psum, const float* __restrict__ pcnt,
                            const float* __restrict__ Wl, const float* __restrict__ bl,
                            float* __restrict__ out, int n_graphs) {
    int tid = blockIdx.x * blockDim.x + threadIdx.x;
    int g = tid >> 1;
    if (g >= n_graphs) return;
    int j = tid & 1;
    float inv = 1.0f / fmaxf(pcnt[g], 1.0f);
    float acc = bl[j];
#pragma unroll
    for (int f = 0; f < NF; ++f)
        acc = fmaf(psum[(size_t)g * NF + f] * inv, Wl[f * 2 + j], acc);
    out[(size_t)g * 2 + j] = acc;
}

static inline int cdiv(long long a, int b) { return (int)((a + b - 1) / b); }

extern "C" void kernel_launch(void* const* d_in, const int* in_sizes, int n_in,
                              void* d_out, int out_size, void* d_ws, size_t ws_size,
                              hipStream_t stream) {
    const float* x     = (const float*)d_in[0];
    const int*   ei    = (const int*)d_in[1];
    const int*   batch = (const int*)d_in[2];
    const float* W1    = (const float*)d_in[3];
    const float* b1    = (const float*)d_in[4];
    const float* W2    = (const float*)d_in[5];
    const float* b2    = (const float*)d_in[6];
    const float* Wl    = (const float*)d_in[7];
    const float* bl    = (const float*)d_in[8];
    float* out = (float*)d_out;

    const int N = in_sizes[0] / 3;   // nodes
    const int E = in_sizes[1] / 2;   // edges
    const int G = out_size / 2;      // graphs
    const int* src = ei;
    const int* dst = ei + E;

    // workspace layout: [dinv | agg | psum | pcnt] (zeroed) | [xw1 | xw2]
    float* ws = (float*)d_ws;
    size_t off = 0;
    float* dinv = ws + off; off += (size_t)N;
    float* agg  = ws + off; off += (size_t)N * NF;
    float* psum = ws + off; off += (size_t)G * NF;
    float* pcnt = ws + off; off += (size_t)G;
    const size_t zero_floats = off;
    float* xw1  = ws + off; off += (size_t)N * NF;
    float* xw2  = ws + off; off += (size_t)N * NF;
    (void)ws_size; (void)n_in;

    const int B = 256;

    // 1) zero accumulators (dinv used as deg accumulator first)
    {
        int n = (int)zero_floats;
        int grid = cdiv(n, B); if (grid > 4096) grid = 4096;
        zero_kernel<<<grid, B, 0, stream>>>(ws, n);
    }
    // 2) in-degree
    degree_kernel<<<cdiv(E, B), B, 0, stream>>>(dst, dinv, E);
    // 3) dinv = rsqrt(deg + 1)
    rsqrt_kernel<<<cdiv(N, B), B, 0, stream>>>(dinv, N);
    // 4) xw1 = x @ W1
    xw1_kernel<<<cdiv((long long)N * 4, B), B, 0, stream>>>(x, W1, xw1, N);
    // 5) layer-1 edge scatter
    scatter_kernel<<<cdiv((long long)E * 8, B), B, 0, stream>>>(src, dst, dinv, xw1, agg, E);
    // 6) h1 = relu(agg + xw1*dinv^2 + b1) (in place in xw1); agg re-zeroed
    finalize1_kernel<<<cdiv((long long)N * 8, B), B, 0, stream>>>(agg, xw1, dinv, b1, N);
    // 7) xw2 = h1 @ W2 (fp32 WMMA)
    {
        int waves = cdiv(N, 16);
        int grid = cdiv(waves, B / 32);
        gemm_wmma_kernel<<<grid, B, 0, stream>>>(xw1, W2, xw2, N);
    }
    // 8) layer-2 edge scatter
    scatter_kernel<<<cdiv((long long)E * 8, B), B, 0, stream>>>(src, dst, dinv, xw2, agg, E);
    // 9) h2 = relu(...) fused with mean-pool accumulation
    finalize2_pool_kernel<<<cdiv((long long)N * 8, B), B, 0, stream>>>(agg, xw2, dinv, b2, batch,
                                                                      psum, pcnt, N);
    // 10) head GEMV
    head_kernel<<<cdiv((long long)G * 2, B), B, 0, stream>>>(psum, pcnt, Wl, bl, out, G);
}